// DeformableConv2d_57827439673367
// MI455X (gfx1250) — compile-verified
//
#include <hip/hip_runtime.h>

typedef float v2f __attribute__((ext_vector_type(2)));
typedef float v8f __attribute__((ext_vector_type(8)));

#define B_  32
#define C_  256
#define H_  224
#define W_  224
#define CO_ 256
#define M_  (B_ * 9)     // 288 rows  (b, pq)
#define N_  CO_          // 256 cols  (co)
#define KD  (C_ * 9)     // 2304 k    (c, uv)

#define S_SZ   (B_ * C_ * 9)   // 73728  shifted[b][c][ij]
#define APK_SZ (KD * M_)       // 663552 A packed [k/2][m][2]
#define WPK_SZ (KD * N_)       // 589824 W packed [k/2][n][2]

// ---------------------------------------------------------------------------
// 1) Bilinear gather: shifted[b,c,i,j] (grid identical for all batches)
// ---------------------------------------------------------------------------
__global__ void gather_kernel(const float* __restrict__ x,
                              const float* __restrict__ offsets,
                              float* __restrict__ S) {
    int t = blockIdx.x * blockDim.x + threadIdx.x;
    if (t >= S_SZ) return;
    int ij = t % 9;
    int bc = t / 9;
    int c  = bc % C_;
    int b  = bc / C_;
    int i  = ij / 3, j = ij % 3;

    float off0 = offsets[ij * 2 + 0];
    float off1 = offsets[ij * 2 + 1];
    // sg = (base + offsets) / (H-1);  grid[...,0]=gx, grid[...,1]=gy
    float gx = ((float)i + off0) / (float)(H_ - 1);
    float gy = ((float)j + off1) / (float)(H_ - 1);
    // align_corners=False unnormalization
    float ix = ((gx + 1.0f) * (float)W_ - 1.0f) * 0.5f;
    float iy = ((gy + 1.0f) * (float)H_ - 1.0f) * 0.5f;

    float x0f = floorf(ix), y0f = floorf(iy);
    float wx1 = ix - x0f, wx0 = 1.0f - wx1;
    float wy1 = iy - y0f, wy0 = 1.0f - wy1;
    int x0 = (int)x0f, y0 = (int)y0f;
    int x1 = x0 + 1,   y1 = y0 + 1;

    const float* img = x + ((size_t)b * C_ + c) * (size_t)(H_ * W_);
    float v00 = (x0 >= 0 && x0 < W_ && y0 >= 0 && y0 < H_) ? img[y0 * W_ + x0] : 0.0f;
    float v10 = (x1 >= 0 && x1 < W_ && y0 >= 0 && y0 < H_) ? img[y0 * W_ + x1] : 0.0f;
    float v01 = (x0 >= 0 && x0 < W_ && y1 >= 0 && y1 < H_) ? img[y1 * W_ + x0] : 0.0f;
    float v11 = (x1 >= 0 && x1 < W_ && y1 >= 0 && y1 < H_) ? img[y1 * W_ + x1] : 0.0f;

    S[t] = v00 * (wx0 * wy0) + v10 * (wx1 * wy0)
         + v01 * (wx0 * wy1) + v11 * (wx1 * wy1);
}

// ---------------------------------------------------------------------------
// 2) Pack weights: Wpk[k>>1][n][k&1] = conv_w[n][k]   (k = c*9 + u*3 + v)
// ---------------------------------------------------------------------------
__global__ void pack_w_kernel(const float* __restrict__ conv_w,
                              float* __restrict__ Wpk) {
    int t = blockIdx.x * blockDim.x + threadIdx.x;
    if (t >= WPK_SZ) return;
    int k2 = t / (N_ * 2);
    int r  = t % (N_ * 2);
    int n  = r >> 1;
    int k  = k2 * 2 + (r & 1);
    Wpk[t] = conv_w[(size_t)n * KD + k];
}

// ---------------------------------------------------------------------------
// 3) im2col w/ zero pad into pair-packed A: Apk[k>>1][m][k&1]
// ---------------------------------------------------------------------------
__global__ void pack_a_kernel(const float* __restrict__ S,
                              float* __restrict__ Apk) {
    int t = blockIdx.x * blockDim.x + threadIdx.x;
    if (t >= APK_SZ) return;
    int k2 = t / (M_ * 2);
    int r  = t % (M_ * 2);
    int m  = r >> 1;
    int k  = k2 * 2 + (r & 1);
    int c  = k / 9, uv = k % 9;
    int u  = uv / 3, v = uv % 3;
    int b  = m / 9, pq = m % 9;
    int p  = pq / 3, q = pq % 3;
    int ii = p + u - 1, jj = q + v - 1;
    float val = 0.0f;
    if (ii >= 0 && ii < 3 && jj >= 0 && jj < 3)
        val = S[b * (C_ * 9) + c * 9 + ii * 3 + jj];
    Apk[t] = val;
}

// ---------------------------------------------------------------------------
// 4) GEMM: D[288x256] = A[288x2304] * W[2304x256], fp32 WMMA 16x16x4.
//    One wave per 16x16 tile. A lane layout (ISA 7.12.2, 32-bit A 16x4):
//    lane = 16*half + l holds A[m = l, k = 2*half + j] in element j.
// ---------------------------------------------------------------------------
__global__ void __launch_bounds__(128) gemm_kernel(const float* __restrict__ Apk,
                                                   const float* __restrict__ Wpk,
                                                   const float* __restrict__ bias,
                                                   float* __restrict__ out) {
    int wave = blockIdx.x * (blockDim.x >> 5) + (threadIdx.x >> 5);
    int lane = threadIdx.x & 31;
    int mt   = wave >> 4;   // 0..17
    int nt   = wave & 15;   // 0..15
    int half = lane >> 4;   // 0,1
    int l    = lane & 15;
    int m    = mt * 16 + l;
    int n    = nt * 16 + l;

    const float* ap = Apk + (size_t)half * (M_ * 2) + (size_t)m * 2;
    const float* bp = Wpk + (size_t)half * (N_ * 2) + (size_t)n * 2;

    v8f acc = {};
#pragma unroll 4
    for (int k2 = 0; k2 < KD / 2; k2 += 2) {   // 576 WMMAs, k-width 4 each
        v2f a = *(const v2f*)ap;
        v2f b = *(const v2f*)bp;
        acc = __builtin_amdgcn_wmma_f32_16x16x4_f32(
            /*neg_a=*/false, a, /*neg_b=*/false, b,
            /*c_mod=*/(short)0, acc, /*reuse_a=*/false, /*reuse_b=*/false);
        ap += M_ * 4;   // advance two k-pairs
        bp += N_ * 4;
    }

    float bv = bias[n];
#pragma unroll
    for (int r = 0; r < 8; ++r) {
        int mm = mt * 16 + r + half * 8;     // C/D layout: VGPR r -> M=r (+8 hi half)
        int b_  = mm / 9;
        int pq  = mm % 9;
        out[b_ * (CO_ * 9) + n * 9 + pq] = acc[r] + bv;
    }
}

// ---------------------------------------------------------------------------
extern "C" void kernel_launch(void* const* d_in, const int* in_sizes, int n_in,
                              void* d_out, int out_size, void* d_ws, size_t ws_size,
                              hipStream_t stream) {
    const float* x       = (const float*)d_in[0];
    const float* offsets = (const float*)d_in[1];
    const float* conv_w  = (const float*)d_in[2];
    const float* conv_b  = (const float*)d_in[3];
    float* ws  = (float*)d_ws;
    float* S   = ws;                 //  73728 f
    float* Apk = S + S_SZ;           // 663552 f
    float* Wpk = Apk + APK_SZ;       // 589824 f  (total ~5.3 MB << ws)
    float* out = (float*)d_out;

    gather_kernel<<<(S_SZ + 255) / 256, 256, 0, stream>>>(x, offsets, S);
    pack_w_kernel<<<(WPK_SZ + 255) / 256, 256, 0, stream>>>(conv_w, Wpk);
    pack_a_kernel<<<(APK_SZ + 255) / 256, 256, 0, stream>>>(S, Apk);
    gemm_kernel<<<(M_ / 16) * (N_ / 16) / 4, 128, 0, stream>>>(Apk, Wpk, conv_b, out);
}